// SoftCutLoss_53369263620650
// MI455X (gfx1250) — compile-verified
//
#include <hip/hip_runtime.h>
#include <hip/hip_bf16.h>

#define B_  16
#define C_  3
#define K_  32
#define H_  256
#define W_  256
#define HW  (H_ * W_)
#define NT  21          // taps with di^2+dj^2 <= 5
#define ND_CHUNKS 8     // x-grid of the WMMA einsum kernel

typedef float v2f __attribute__((ext_vector_type(2)));
typedef float v8f __attribute__((ext_vector_type(8)));

// ---- tap table: (di,dj) with di^2+dj^2 <= 5, weight = exp(-d2/16) ----
__constant__ int   c_di[NT] = { -2,-2,-2,  -1,-1,-1,-1,-1,  0, 0, 0, 0, 0,  1, 1, 1, 1, 1,  2, 2, 2 };
__constant__ int   c_dj[NT] = { -1, 0, 1,  -2,-1, 0, 1, 2, -2,-1, 0, 1, 2, -2,-1, 0, 1, 2, -1, 0, 1 };
__constant__ float c_dw[NT] = {
    0.73161562894664711f, 0.77880078307140487f, 0.73161562894664711f,            // d2=5,4,5
    0.73161562894664711f, 0.88249690258459546f, 0.93941306281347578f,            // d2=5,2,1
    0.88249690258459546f, 0.73161562894664711f,                                  // d2=2,5
    0.77880078307140487f, 0.93941306281347578f, 1.0f,                            // d2=4,1,0
    0.93941306281347578f, 0.77880078307140487f,                                  // d2=1,4
    0.73161562894664711f, 0.88249690258459546f, 0.93941306281347578f,            // d2=5,2,1
    0.88249690258459546f, 0.73161562894664711f,                                  // d2=2,5
    0.73161562894664711f, 0.77880078307140487f, 0.73161562894664711f };          // d2=5,4,5

// ---------------- pass 1: gray image = mean over 3 channels ----------------
__global__ __launch_bounds__(256) void img_kernel(const float* __restrict__ batch,
                                                  float* __restrict__ img) {
    int i = blockIdx.x * 256 + threadIdx.x;          // i indexes (b, p)
    int b = i / HW;
    int p = i - b * HW;
    const float* base = batch + (size_t)b * C_ * HW + p;
    img[i] = (base[0] + base[HW] + base[2 * HW]) * (1.0f / 3.0f);
}

// ------- pass 2: bilateral stencil, accumulate T(K,H,W), U(H,W) over b -------
__global__ __launch_bounds__(256) void tu_kernel(const float* __restrict__ enc,
                                                 const float* __restrict__ img,
                                                 float* __restrict__ T,
                                                 float* __restrict__ U) {
    __shared__ float simg[20 * 20];
    __shared__ float senc[20 * 20];
    const int lx = threadIdx.x & 15, ly = threadIdx.x >> 4;
    const int x0 = blockIdx.x * 16, y0 = blockIdx.y * 16;
    const int gx = x0 + lx, gy = y0 + ly;

    float acc[K_];
    #pragma unroll
    for (int k = 0; k < K_; ++k) acc[k] = 0.0f;
    float uacc = 0.0f;

    for (int b = 0; b < B_; ++b) {
        // stage 20x20 img tile (2-halo, zero-padded like the reference)
        __syncthreads();
        for (int idx = threadIdx.x; idx < 400; idx += 256) {
            int yy = y0 - 2 + idx / 20;
            int xx = x0 - 2 + idx % 20;
            float v = 0.0f;
            if (yy >= 0 && yy < H_ && xx >= 0 && xx < W_) v = img[(size_t)b * HW + yy * W_ + xx];
            simg[idx] = v;
        }
        __syncthreads();

        const float cpix = simg[(ly + 2) * 20 + lx + 2];
        float w[NT];
        #pragma unroll
        for (int t = 0; t < NT; ++t) {
            float nb  = simg[(ly + 2 + c_di[t]) * 20 + (lx + 2 + c_dj[t])];
            float dlt = nb - cpix;
            w[t] = __expf(-dlt * dlt * 0.01f) * c_dw[t];   // oi^2 = 100
            uacc += w[t];
        }

        const float* encb = enc + (size_t)b * K_ * HW;
        for (int k = 0; k < K_; ++k) {
            if (k + 1 < K_)   // gfx1250: global_prefetch_b8 for the next channel slice
                __builtin_prefetch(encb + (size_t)(k + 1) * HW + gy * W_ + gx, 0, 1);
            __syncthreads();
            for (int idx = threadIdx.x; idx < 400; idx += 256) {
                int yy = y0 - 2 + idx / 20;
                int xx = x0 - 2 + idx % 20;
                float v = 0.0f;
                if (yy >= 0 && yy < H_ && xx >= 0 && xx < W_) v = encb[(size_t)k * HW + yy * W_ + xx];
                senc[idx] = v;
            }
            __syncthreads();
            float s = 0.0f;
            #pragma unroll
            for (int t = 0; t < NT; ++t)
                s = fmaf(w[t], senc[(ly + 2 + c_di[t]) * 20 + (lx + 2 + c_dj[t])], s);
            acc[k] += s;
        }
    }
    const int p = gy * W_ + gx;
    #pragma unroll
    for (int k = 0; k < K_; ++k) T[(size_t)k * HW + p] = acc[k];
    U[p] = uacc;
}

// ---------------- pass 3a: zero the N/D accumulators ----------------
__global__ __launch_bounds__(256) void zero_kernel(float* __restrict__ p) {
    p[blockIdx.x * 256 + threadIdx.x] = 0.0f;
}

// ------- pass 3b: N[k,b] = <enc[b,k,:], T[k,:]>, D[k,b] = <enc[b,k,:], U> via WMMA f32 16x16x4 -------
// C columns: n=0 accumulates N, n=1 accumulates D. Columns n>=2 accumulate garbage
// (lanes n>=2 just re-load U) — harmless because column n of D depends only on
// column n of B, and we never read those columns. This keeps the loop branchless.
__global__ __launch_bounds__(256) void nd_kernel(const float* __restrict__ enc,
                                                 const float* __restrict__ T,
                                                 const float* __restrict__ U,
                                                 float* __restrict__ Nbuf,
                                                 float* __restrict__ Dbuf) {
    const int k    = blockIdx.y;
    const int wave = threadIdx.x >> 5;       // 0..7
    const int lane = threadIdx.x & 31;
    const int n    = lane & 15;              // A row (batch) == B/C column index
    const int half = lane >> 4;              // K-pair selector
    const int POS_PER_WG   = HW / ND_CHUNKS;     // 8192
    const int POS_PER_WAVE = POS_PER_WG / 8;     // 1024
    const int p0 = blockIdx.x * POS_PER_WG + wave * POS_PER_WAVE;

    // A-matrix row m = lane&15 -> encoded[b=m, k, :]
    const float* enck = enc + (size_t)n * K_ * HW + (size_t)k * HW;
    const float* Tk   = T + (size_t)k * HW;
    // per-lane B base pointer, chosen once (branchless inner loop)
    const float* bsrc = (n == 0) ? Tk : U;

    v8f c0 = {0.f, 0.f, 0.f, 0.f, 0.f, 0.f, 0.f, 0.f};
    v8f c1 = {0.f, 0.f, 0.f, 0.f, 0.f, 0.f, 0.f, 0.f};
    const int kk = half * 2;                 // this lane carries K=kk, kk+1
    #pragma unroll 2
    for (int p = p0; p < p0 + POS_PER_WAVE; p += 8) {
        v2f a0, b0, a1, b1;
        a0.x = enck[p + kk];     a0.y = enck[p + kk + 1];
        b0.x = bsrc[p + kk];     b0.y = bsrc[p + kk + 1];
        a1.x = enck[p + 4 + kk]; a1.y = enck[p + 4 + kk + 1];
        b1.x = bsrc[p + 4 + kk]; b1.y = bsrc[p + 4 + kk + 1];
        c0 = __builtin_amdgcn_wmma_f32_16x16x4_f32(false, a0, false, b0, (short)0, c0, false, false);
        c1 = __builtin_amdgcn_wmma_f32_16x16x4_f32(false, a1, false, b1, (short)0, c1, false, false);
    }

    // C element (M = v + 8*half, N = lane&15) lives in VGPR v; flush columns 0/1
    if (n < 2) {
        float* dst = (n == 0) ? Nbuf : Dbuf;
        #pragma unroll
        for (int v = 0; v < 8; ++v) {
            int m = v + 8 * half;            // batch index
            atomicAdd(&dst[k * 16 + m], c0[v] + c1[v]);
        }
    }
}

// ---------------- pass 4: loss = mean_b (K - sum_k N/D) ----------------
__global__ void final_kernel(const float* __restrict__ Nbuf,
                             const float* __restrict__ Dbuf,
                             float* __restrict__ out) {
    if (threadIdx.x == 0 && blockIdx.x == 0) {
        float total = 0.0f;
        for (int b = 0; b < B_; ++b) {
            float s = 0.0f;
            for (int k = 0; k < K_; ++k) s += Nbuf[k * 16 + b] / Dbuf[k * 16 + b];
            total += ((float)K_ - s);
        }
        out[0] = total / (float)B_;
    }
}

extern "C" void kernel_launch(void* const* d_in, const int* in_sizes, int n_in,
                              void* d_out, int out_size, void* d_ws, size_t ws_size,
                              hipStream_t stream) {
    const float* batch = (const float*)d_in[0];   // (16,3,256,256)
    const float* enc   = (const float*)d_in[1];   // (16,32,256,256)
    float* out = (float*)d_out;

    float* ws  = (float*)d_ws;
    float* img = ws;                        // 16*HW      = 4 MB
    float* T   = img + (size_t)B_ * HW;     // 32*HW      = 8 MB
    float* U   = T + (size_t)K_ * HW;       // HW         = 256 KB
    float* Nb  = U + HW;                    // 512 floats
    float* Db  = Nb + K_ * B_;              // 512 floats

    img_kernel<<<(B_ * HW) / 256, 256, 0, stream>>>(batch, img);
    tu_kernel<<<dim3(W_ / 16, H_ / 16), 256, 0, stream>>>(enc, img, T, U);
    zero_kernel<<<(2 * K_ * B_) / 256, 256, 0, stream>>>(Nb);   // zeros Nb and Db (contiguous)
    nd_kernel<<<dim3(ND_CHUNKS, K_), 256, 0, stream>>>(enc, T, U, Nb, Db);
    final_kernel<<<1, 32, 0, stream>>>(Nb, Db, out);
}